// IMPACTModel_21234318311841
// MI455X (gfx1250) — compile-verified
//
#include <hip/hip_runtime.h>

// IMPACT model modality-argmin kernel for MI455X (gfx1250).
//
// Per query b: p[m] = || u[user_b] - e[item_b, m] ||^2  for m = 1..nb(item_b)
//              out[b] = (argmin_m p[m] - 1) / (nb - 1) + 1
//
// Bandwidth-bound gather op (~0.04 flop/byte; unique working set ~194 MB vs
// 192 MB L2 -> L2-gather-bound). Strategy:
//  - one wave32 per query; lane owns 4 of the 128 concepts (one float4 / b128)
//  - item rows m=1..nb gathered via CDNA5 GLOBAL_LOAD_ASYNC_TO_LDS_B128
//    (ASYNCcnt path) into a per-wave LDS slice, overlapped with the user load
//  - mask[] input never touched: mask == 0 iff 1 <= m <= nb, +inf else, so
//    the m-loop bound nb reproduces it exactly and halves item gather traffic
//  - wave index + ids forced onto the scalar path (readfirstlane) so the id
//    fetches are s_load_b32 and all row-base math is SALU
//  - 5-step __shfl_xor butterfly for the 32-lane reduction, argmin in regs

#define MOD_M        14                  // NB_MOD_MAX + 2
#define CONCEPTS     128
#define ROW_BYTES    (CONCEPTS * 4)      // 512 B per embedding row
#define MAX_ROWS     12                  // nb in [2, 12] -> rows m = 1..12
#define WAVE_LDS     (MAX_ROWS * ROW_BYTES)        // 6144 B per wave
#define WAVES_PER_BLOCK 8
#define BLOCK_LDS    (WAVES_PER_BLOCK * WAVE_LDS)  // 49152 B

__global__ __launch_bounds__(WAVES_PER_BLOCK * 32)
void impact_mod_to_resp_kernel(const int*   __restrict__ user_ids,
                               const int*   __restrict__ item_ids,
                               const float* __restrict__ users_emb,   // [USER_N,128]
                               const float* __restrict__ item_emb,    // [ITEM_N*14,128]
                               const int*   __restrict__ nb_mod,      // [ITEM_N]
                               float*       __restrict__ out,
                               int B)
{
    extern __shared__ char smem[];

    const int lane = threadIdx.x & 31;
    // Wave index is uniform within the wave: pin it (and therefore b, the id
    // loads, and all row-base math) to the scalar pipe.
    const int wib  = __builtin_amdgcn_readfirstlane((int)(threadIdx.x >> 5));
    const int b    = blockIdx.x * WAVES_PER_BLOCK + wib;     // scalar
    if (b >= B) return;

    const int user = __builtin_amdgcn_readfirstlane(user_ids[b]);  // s_load
    const int item = __builtin_amdgcn_readfirstlane(item_ids[b]);  // s_load
    const int nb   = __builtin_amdgcn_readfirstlane(nb_mod[item]); // s_load, [2,12]

    // This wave's private LDS slice. Generic LDS pointers carry the LDS byte
    // offset in their low 32 bits (aperture encoding), which is exactly what
    // the async-to-LDS VDST VGPR wants.
    char* wbase = smem + wib * WAVE_LDS;
    const unsigned lds_lane =
        (unsigned)(uintptr_t)wbase + (unsigned)lane * 16u;

    // ---- Fire all item-row gathers down the async Global->LDS pipe --------
    // GVS addressing: mem = SADDR(64b base) + VADDR(32b per-lane offset).
    const unsigned long long ibase64 = (unsigned long long)(uintptr_t)item_emb;
    const unsigned row0 = (unsigned)((item * MOD_M + 1) * ROW_BYTES); // scalar
    unsigned voff = row0 + (unsigned)lane * 16u;
    #pragma unroll 1
    for (int m = 1; m <= nb; ++m) {
        unsigned ldsa = lds_lane + (unsigned)(m - 1) * (unsigned)ROW_BYTES;
        asm volatile(
            "global_load_async_to_lds_b128 %0, %1, %2 offset:0"
            :
            : "v"(ldsa), "v"(voff), "s"(ibase64)
            : "memory");
        voff += (unsigned)ROW_BYTES;
    }

    // ---- Overlap: pull this wave's user row into VGPRs while DMA runs -----
    const float4 u4 =
        *(const float4*)(users_emb + (size_t)user * CONCEPTS + (size_t)lane * 4);

    // Drain the async counter: all rows are now resident in our LDS slice.
    asm volatile("s_wait_asynccnt 0x0" ::: "memory");

    // ---- Distances + running argmin ---------------------------------------
    const float4* rows = (const float4*)wbase;               // LDS, ds_load_b128
    float best  = __builtin_inff();
    int   bestm = 1;
    #pragma unroll 1
    for (int m = 1; m <= nb; ++m) {
        float4 e4 = rows[(m - 1) * 32 + lane];
        float dx = u4.x - e4.x;
        float dy = u4.y - e4.y;
        float dz = u4.z - e4.z;
        float dw = u4.w - e4.w;
        float s = dx * dx + dy * dy + dz * dz + dw * dw;
        // 32-lane butterfly sum (swizzle/permlane on gfx1250 wave32)
        s += __shfl_xor(s, 16, 32);
        s += __shfl_xor(s,  8, 32);
        s += __shfl_xor(s,  4, 32);
        s += __shfl_xor(s,  2, 32);
        s += __shfl_xor(s,  1, 32);
        if (s < best) { best = s; bestm = m; }               // first-min: matches jnp.argmin
    }

    if (lane == 0)
        out[b] = (float)(bestm - 1) / (float)(nb - 1) + 1.0f;
}

extern "C" void kernel_launch(void* const* d_in, const int* in_sizes, int n_in,
                              void* d_out, int out_size, void* d_ws, size_t ws_size,
                              hipStream_t stream)
{
    // setup_inputs() order:
    //  0 user_ids[B] i32, 1 item_ids[B] i32, 2 concept_ids[B] i32 (unused),
    //  3 users_emb_weight[USER_N*128] f32, 4 item_resp_emb_weight[ITEM_N*14*128] f32,
    //  5 mask[ITEM_N*14] f32 (provably redundant given nb_modalities -> unused),
    //  6 nb_modalities[ITEM_N] i32
    const int*   user_ids  = (const int*)  d_in[0];
    const int*   item_ids  = (const int*)  d_in[1];
    const float* users_emb = (const float*)d_in[3];
    const float* item_emb  = (const float*)d_in[4];
    const int*   nb_mod    = (const int*)  d_in[6];
    float*       out       = (float*)      d_out;

    const int B = in_sizes[0];
    const int block = WAVES_PER_BLOCK * 32;                  // 256 threads = 8 waves
    const int grid  = (B + WAVES_PER_BLOCK - 1) / WAVES_PER_BLOCK;

    impact_mod_to_resp_kernel<<<grid, block, BLOCK_LDS, stream>>>(
        user_ids, item_ids, users_emb, item_emb, nb_mod, out, B);
}